// Gridnet_9457517986054
// MI455X (gfx1250) — compile-verified
//
#include <hip/hip_runtime.h>
#include <math.h>

#define GRID  64
#define BS    8
#define NB    8
#define ITERS 8
#define BATCH 16
#define EPS   1e-5f
#define TILE3 1000          // (8+2)^3 real elements per batch
#define TSTR  1024          // padded per-batch tile stride (floats); pad holds zeros
#define CELLS 512           // 8^3
#define NTHREADS 512        // thread == cell; 16 waves

// Low 32 bits of a generic pointer to __shared__ == LDS byte offset
#define LDS_OFF(p) ((unsigned)(unsigned long long)(p))

__global__ __launch_bounds__(NTHREADS)
void gridnet_kernel(const float* __restrict__ x,
                    const float* __restrict__ weight,
                    const float* __restrict__ bias,
                    float* __restrict__ out)
{
    __shared__ alignas(16) float tile[BATCH * TSTR];   // 64 KB: raw activations + frozen halo
    __shared__ alignas(16) float stats[BATCH * 2];     // {rstd, mean*rstd} per batch

    const int tid = threadIdx.x;
    const int blk = blockIdx.x;              // 0..511
    const int bx = blk >> 6, by = (blk >> 3) & 7, bz = blk & 7;
    const int gx0 = bx << 3, gy0 = by << 3, gz0 = bz << 3;

    // this thread's cell (fixed for the whole kernel)
    const int c   = tid;
    const int lx  = c >> 6, ly = (c >> 3) & 7, lz = c & 7;
    const int tb0 = lx * 100 + ly * 10 + lz;                // tap (di,dj,dk) at +di*100+dj*10+dk
    const size_t gcell = ((size_t)(gx0 + lx) << 12) + ((gy0 + ly) << 6) + (gz0 + lz);

    // ---------- tile: async DMA for in-bounds elements, ds-store zeros for OOB halo + pad ----------
    for (int idx = tid; idx < BATCH * TSTR; idx += NTHREADS) {   // 32 per thread
        int b = idx >> 10;
        int r = idx & (TSTR - 1);
        bool inb = false;
        if (r < TILE3) {
            int tx = r / 100;
            int r2 = r - tx * 100;
            int ty = r2 / 10;
            int tz = r2 - ty * 10;
            int gx = gx0 - 1 + tx, gy = gy0 - 1 + ty, gz = gz0 - 1 + tz;
            if ((unsigned)gx < GRID && (unsigned)gy < GRID && (unsigned)gz < GRID) {
                inb = true;
                const float* src = x + ((size_t)b << 18) + ((size_t)gx << 12) + (gy << 6) + gz;
                unsigned dst = LDS_OFF(&tile[idx]);
                asm volatile("global_load_async_to_lds_b32 %0, %1, off"
                             :: "v"(dst), "v"(src) : "memory");
            }
        }
        if (!inb) tile[idx] = 0.0f;   // disjoint addresses from the async writes: no race
    }

    // ---------- weights + bias -> registers, pinned for all 8 iterations ----------
    float wreg[27];
    #pragma unroll
    for (int w = 0; w < 27; ++w)
        wreg[w] = weight[((size_t)w << 18) + gcell];
    float breg = bias[gcell];

    // per-cell constant: sum of the 27 weights (for factored normalization)
    float wsum = 0.0f;
    #pragma unroll
    for (int w = 0; w < 27; ++w) wsum += wreg[w];

    asm volatile("s_wait_asynccnt 0" ::: "memory");
    __syncthreads();

    const int jw = tid & 31;     // lane in wave
    const int bw = tid >> 5;     // wave id == batch for the stats pass
    const float inv = 1.0f / (float)TILE3;

    #pragma unroll 1
    for (int it = 0; it < ITERS; ++it) {
        // ---- stats for batch bw over its 10^3 tile (zeros in pad don't affect sum/sumsq) ----
        float s = 0.0f, qs = 0.0f;
        #pragma unroll
        for (int m = 0; m < 8; ++m) {
            const float4 v4 = *(const float4*)&tile[bw * TSTR + m * 128 + jw * 4];
            s  += (v4.x + v4.y) + (v4.z + v4.w);
            qs += (v4.x * v4.x + v4.y * v4.y) + (v4.z * v4.z + v4.w * v4.w);
        }
        #pragma unroll
        for (int m = 1; m < 32; m <<= 1) {
            s  += __shfl_xor(s,  m, 32);
            qs += __shfl_xor(qs, m, 32);
        }
        if (jw == 0) {
            float mean = s * inv;
            float var  = qs * inv - mean * mean;
            float rstd = rsqrtf(var + EPS);
            stats[2 * bw]     = rstd;
            stats[2 * bw + 1] = mean * rstd;    // nm: norm(v) = v*rstd - nm
        }
        __syncthreads();

        // ---- 27-tap stencil for this cell, all 16 batches ----
        // Σ_w (nv*rstd - nm)·W_w + bias  ==  rstd·(Σ_w nv·W_w) + (bias - nm·Σ_w W_w)
        float newv[BATCH];
        #pragma unroll
        for (int b = 0; b < BATCH; ++b) {
            const int tb = b * TSTR + tb0;
            float acc = 0.0f;
            #pragma unroll
            for (int di = 0; di < 3; ++di)
                #pragma unroll
                for (int dj = 0; dj < 3; ++dj)
                    #pragma unroll
                    for (int dk = 0; dk < 3; ++dk)
                        acc = __builtin_fmaf(tile[tb + di * 100 + dj * 10 + dk],
                                             wreg[(di * 3 + dj) * 3 + dk], acc);
            float rstd = stats[2 * b];
            float nm   = stats[2 * b + 1];
            float d    = __builtin_fmaf(rstd, acc, __builtin_fmaf(-nm, wsum, breg));
            newv[b] = d * (1.0f / (1.0f + __expf(-d)));   // SiLU
        }

        __syncthreads();   // all tile reads done before any interior write
        #pragma unroll
        for (int b = 0; b < BATCH; ++b)
            tile[b * TSTR + tb0 + 111] = newv[b];   // (lx+1,ly+1,lz+1) = tb0 + 111
        __syncthreads();
    }

    // ---------- async store: interior LDS -> out ----------
    #pragma unroll 1
    for (int b = 0; b < BATCH; ++b) {
        unsigned src = LDS_OFF(&tile[b * TSTR + tb0 + 111]);
        const float* dst = out + ((size_t)b << 18) + gcell;
        asm volatile("global_store_async_from_lds_b32 %0, %1, off"
                     :: "v"(dst), "v"(src) : "memory");
    }
    asm volatile("s_wait_asynccnt 0" ::: "memory");
}

extern "C" void kernel_launch(void* const* d_in, const int* in_sizes, int n_in,
                              void* d_out, int out_size, void* d_ws, size_t ws_size,
                              hipStream_t stream) {
    const float* x    = (const float*)d_in[0];   // [16,64,64,64] f32
    const float* w    = (const float*)d_in[1];   // [27,64,64,64] f32
    const float* bias = (const float*)d_in[2];   // [64,64,64]    f32
    float* out = (float*)d_out;                  // [16,64,64,64] f32
    (void)in_sizes; (void)n_in; (void)out_size; (void)d_ws; (void)ws_size;
    gridnet_kernel<<<dim3(NB * NB * NB), dim3(NTHREADS), 0, stream>>>(x, w, bias, out);
}